// TGCNCell_81363860455818
// MI455X (gfx1250) — compile-verified
//
#include <hip/hip_runtime.h>
#include <cstdint>

#define NN  10000
#define EE  320000
#define INC 128
#define HH  256

typedef __attribute__((ext_vector_type(2))) float v2f;
typedef __attribute__((ext_vector_type(8))) float v8f;

// ---------------- degree / normalization ----------------
__global__ void deg_init(float* __restrict__ deg) {
    int i = blockIdx.x * blockDim.x + threadIdx.x;
    if (i < NN) deg[i] = 1.0f;          // self loop
}

__global__ void deg_acc(const long long* __restrict__ dst, float* __restrict__ deg) {
    int e = blockIdx.x * blockDim.x + threadIdx.x;
    if (e < EE) atomicAdd(&deg[(int)dst[e]], 1.0f);
}

__global__ void deg_rsqrt(float* __restrict__ deg) {
    int i = blockIdx.x * blockDim.x + threadIdx.x;
    if (i < NN) deg[i] = rsqrtf(deg[i]);
}

// ---------------- WMMA f32 GEMM: C[M,256] (+)= A[M,K] @ W[K,256] ----------------
// grid.x = M/16 row tiles; block = 128 threads = 4 waves; wave w owns cols [64w,64w+64).
template<bool ACC>
__global__ __launch_bounds__(128) void wmma_gemm(const float* __restrict__ A,
                                                 const float* __restrict__ W,
                                                 float* __restrict__ C, int K) {
    const int lane = threadIdx.x & 31;
    const int wave = threadIdx.x >> 5;
    const int lh   = lane >> 4;          // 0: lanes 0-15, 1: lanes 16-31
    const int li   = lane & 15;
    const long row  = (long)blockIdx.x * 16;
    const int  col0 = wave * 64;

    v8f acc[4] = {};
    const float* arow = A + (row + li) * (long)K + lh * 2;   // M=li, K pair by half
    for (int k = 0; k < K; k += 4) {
        v2f a = *(const v2f*)(arow + k);                     // A[M=li][k+2lh .. +1]
        const float* wbase = W + (long)(k + lh * 2) * HH + col0 + li;
#pragma unroll
        for (int t = 0; t < 4; ++t) {
            v2f b;
            b.x = wbase[t * 16];                              // W[k+2lh  ][n]
            b.y = wbase[t * 16 + HH];                         // W[k+2lh+1][n]
            acc[t] = __builtin_amdgcn_wmma_f32_16x16x4_f32(
                false, a, false, b, (short)0, acc[t], false, false);
        }
    }
#pragma unroll
    for (int t = 0; t < 4; ++t) {
#pragma unroll
        for (int j = 0; j < 8; ++j) {
            long r = row + j + lh * 8;                        // C/D layout: M=j+8*half
            int  c = col0 + t * 16 + li;
            float v = acc[t][j];
            if (ACC) v += C[r * HH + c];
            C[r * HH + c] = v;
        }
    }
}

// ---------------- self-loop init: agg = xw * dinv[i]^2 ----------------
__global__ __launch_bounds__(256) void selfloop(const float* __restrict__ xw,
                                                const float* __restrict__ dinv,
                                                float* __restrict__ agg) {
    long idx = ((long)blockIdx.x * blockDim.x + threadIdx.x) * 4;
    long i = idx / HH;
    float s = dinv[i]; s = s * s;
    float4 v = *(const float4*)(xw + idx);
    float4 o = {v.x * s, v.y * s, v.z * s, v.w * s};
    *(float4*)(agg + idx) = o;
}

// ---------------- edge scatter: agg[dst] += xw[src] * dinv[src]*dinv[dst] ----------------
// 4 edges per 256-thread block; 64 lanes * float4 per edge row.
__global__ __launch_bounds__(256) void gcn_scatter(const long long* __restrict__ src,
                                                   const long long* __restrict__ dst,
                                                   const float* __restrict__ dinv,
                                                   const float* __restrict__ xw,
                                                   float* __restrict__ agg) {
    const int t = threadIdx.x;
    const long e = (long)blockIdx.x * 4 + (t >> 6);
    const int  j = (t & 63) * 4;
    const long s = (long)src[e];
    const long d = (long)dst[e];
    const float nrm = dinv[s] * dinv[d];
    const float4 v = *(const float4*)(xw + s * HH + j);
    float* p = agg + d * HH + j;
    atomicAdd(p + 0, v.x * nrm);
    atomicAdd(p + 1, v.y * nrm);
    atomicAdd(p + 2, v.z * nrm);
    atomicAdd(p + 3, v.w * nrm);
}

// ---------------- elementwise epilogues ----------------
__device__ __forceinline__ float sigf(float x) { return 1.0f / (1.0f + expf(-x)); }

template<int MODE>   // 0 = relu, 1 = tanh
__global__ __launch_bounds__(256) void bias_act(float* __restrict__ x,
                                                const float* __restrict__ b) {
    long idx = ((long)blockIdx.x * blockDim.x + threadIdx.x) * 4;
    int j = (int)(idx % HH);
    float4 bb = *(const float4*)(b + j);
    float4 v = *(float4*)(x + idx);
    v.x += bb.x; v.y += bb.y; v.z += bb.z; v.w += bb.w;
    if (MODE == 0) {
        v.x = fmaxf(v.x, 0.f); v.y = fmaxf(v.y, 0.f);
        v.z = fmaxf(v.z, 0.f); v.w = fmaxf(v.w, 0.f);
    } else {
        v.x = tanhf(v.x); v.y = tanhf(v.y); v.z = tanhf(v.z); v.w = tanhf(v.w);
    }
    *(float4*)(x + idx) = v;
}

__global__ __launch_bounds__(256) void bias_sigmoid(float* __restrict__ x,
                                                    const float* __restrict__ b) {
    long idx = ((long)blockIdx.x * blockDim.x + threadIdx.x) * 4;
    int j = (int)(idx % HH);
    float4 bb = *(const float4*)(b + j);
    float4 v = *(float4*)(x + idx);
    v.x = sigf(v.x + bb.x); v.y = sigf(v.y + bb.y);
    v.z = sigf(v.z + bb.z); v.w = sigf(v.w + bb.w);
    *(float4*)(x + idx) = v;
}

// r := sigmoid(r + b) * h_prev   (in place)
__global__ __launch_bounds__(256) void rh_fuse(float* __restrict__ r,
                                               const float* __restrict__ b,
                                               const float* __restrict__ hp) {
    long idx = ((long)blockIdx.x * blockDim.x + threadIdx.x) * 4;
    int j = (int)(idx % HH);
    float4 bb = *(const float4*)(b + j);
    float4 v = *(float4*)(r + idx);
    float4 h = *(const float4*)(hp + idx);
    v.x = sigf(v.x + bb.x) * h.x; v.y = sigf(v.y + bb.y) * h.y;
    v.z = sigf(v.z + bb.z) * h.z; v.w = sigf(v.w + bb.w) * h.w;
    *(float4*)(r + idx) = v;
}

// out = u*h_prev + (1-u)*tanh(cpre + b)
__global__ __launch_bounds__(256) void final_fuse(const float* __restrict__ cpre,
                                                  const float* __restrict__ b,
                                                  const float* __restrict__ u,
                                                  const float* __restrict__ hp,
                                                  float* __restrict__ out) {
    long idx = ((long)blockIdx.x * blockDim.x + threadIdx.x) * 4;
    int j = (int)(idx % HH);
    float4 bb = *(const float4*)(b + j);
    float4 c = *(const float4*)(cpre + idx);
    float4 uu = *(const float4*)(u + idx);
    float4 h = *(const float4*)(hp + idx);
    float4 o;
    o.x = uu.x * h.x + (1.f - uu.x) * tanhf(c.x + bb.x);
    o.y = uu.y * h.y + (1.f - uu.y) * tanhf(c.y + bb.y);
    o.z = uu.z * h.z + (1.f - uu.z) * tanhf(c.z + bb.z);
    o.w = uu.w * h.w + (1.f - uu.w) * tanhf(c.w + bb.w);
    *(float4*)(out + idx) = o;
}

// ---------------- orchestration ----------------
extern "C" void kernel_launch(void* const* d_in, const int* in_sizes, int n_in,
                              void* d_out, int out_size, void* d_ws, size_t ws_size,
                              hipStream_t stream) {
    const float*     x_t  = (const float*)d_in[0];
    const float*     hp   = (const float*)d_in[1];
    const long long* ei   = (const long long*)d_in[2];   // [2, E] int64
    const float*     g1w  = (const float*)d_in[3];
    const float*     g1b  = (const float*)d_in[4];
    const float*     g2w  = (const float*)d_in[5];
    const float*     g2b  = (const float*)d_in[6];
    const float*     Wu   = (const float*)d_in[7];
    const float*     bu   = (const float*)d_in[8];
    const float*     Wr   = (const float*)d_in[9];
    const float*     br   = (const float*)d_in[10];
    const float*     Wc   = (const float*)d_in[11];
    const float*     bc   = (const float*)d_in[12];
    float* out = (float*)d_out;

    const long long* src = ei;
    const long long* dst = ei + EE;

    const size_t NH = (size_t)NN * HH;
    float* dinv = (float*)d_ws;          // NN floats (padded to 16384)
    float* buf0 = dinv + 16384;          // xw1  -> u
    float* buf1 = buf0 + NH;             // h    -> r -> r*h_prev
    float* buf2 = buf1 + NH;             // xw2  -> c_pre
    float* buf3 = buf2 + NH;             // z_t

    const int EW = 2500;                 // (N*H/4)/256 elementwise blocks

    // symmetric normalization
    deg_init <<<(NN + 255) / 256, 256, 0, stream>>>(dinv);
    deg_acc  <<<(EE + 255) / 256, 256, 0, stream>>>(dst, dinv);
    deg_rsqrt<<<(NN + 255) / 256, 256, 0, stream>>>(dinv);

    // GCN layer 1: h = relu(Dinv(A+I)Dinv @ (x W1) + b1)
    wmma_gemm<false><<<NN / 16, 128, 0, stream>>>(x_t, g1w, buf0, INC);
    selfloop   <<<EW, 256, 0, stream>>>(buf0, dinv, buf1);
    gcn_scatter<<<EE / 4, 256, 0, stream>>>(src, dst, dinv, buf0, buf1);
    bias_act<0><<<EW, 256, 0, stream>>>(buf1, g1b);

    // GCN layer 2: z = tanh(Dinv(A+I)Dinv @ (h W2) + b2)
    wmma_gemm<false><<<NN / 16, 128, 0, stream>>>(buf1, g2w, buf2, HH);
    selfloop   <<<EW, 256, 0, stream>>>(buf2, dinv, buf3);
    gcn_scatter<<<EE / 4, 256, 0, stream>>>(src, dst, dinv, buf2, buf3);
    bias_act<1><<<EW, 256, 0, stream>>>(buf3, g2b);

    // u = sigmoid([z, h_prev] @ Wu + bu)  (split-K concat)
    wmma_gemm<false><<<NN / 16, 128, 0, stream>>>(buf3, Wu, buf0, HH);
    wmma_gemm<true> <<<NN / 16, 128, 0, stream>>>(hp, Wu + (size_t)HH * HH, buf0, HH);
    bias_sigmoid<<<EW, 256, 0, stream>>>(buf0, bu);

    // r*h_prev = sigmoid([z, h_prev] @ Wr + br) * h_prev
    wmma_gemm<false><<<NN / 16, 128, 0, stream>>>(buf3, Wr, buf1, HH);
    wmma_gemm<true> <<<NN / 16, 128, 0, stream>>>(hp, Wr + (size_t)HH * HH, buf1, HH);
    rh_fuse<<<EW, 256, 0, stream>>>(buf1, br, hp);

    // c_pre = [z, r*h_prev] @ Wc ; out = u*h_prev + (1-u)*tanh(c_pre + bc)
    wmma_gemm<false><<<NN / 16, 128, 0, stream>>>(buf3, Wc, buf2, HH);
    wmma_gemm<true> <<<NN / 16, 128, 0, stream>>>(buf1, Wc + (size_t)HH * HH, buf2, HH);
    final_fuse<<<EW, 256, 0, stream>>>(buf2, bc, buf0, hp, out);
}